// dropping_noise_58858231824551
// MI455X (gfx1250) — compile-verified
//
#include <hip/hip_runtime.h>
#include <hip/hip_bf16.h>
#include <math.h>

typedef float v2f __attribute__((ext_vector_type(2)));
typedef float v4f __attribute__((ext_vector_type(4)));
typedef float v8f __attribute__((ext_vector_type(8)));

__constant__ float DEC_LO[16] = {
  -0.0033824159510061256f, -0.0005421323317911481f, 0.03169508781149298f,
   0.007607487324917605f,  -0.1432942383508097f,   -0.061273359067658524f,
   0.4813596512583722f,     0.7771857517005235f,    0.3644418948353314f,
  -0.05194583810770904f,   -0.027219029917056003f,  0.049137179673607506f,
   0.003808752013890615f,  -0.01495225833704823f,  -0.0003029205147213668f,
   0.0018899503327594609f };

// symmetric (half-sample) extension index, single reflection is enough here
__device__ __forceinline__ int symidx(int i, int n) {
  if (i < 0) i = -i - 1;
  if (i >= n) i = 2 * n - 1 - i;
  return i;
}

// fp32 WMMA: D(16x16) += A(16x4) * B(4x16), exact fp32 accumulate
static __device__ __forceinline__ v8f wmma4(v2f a, v2f b, v8f c) {
  return __builtin_amdgcn_wmma_f32_16x16x4_f32(false, a, false, b, (short)0, c,
                                               false, false);
}

// ---------------------------------------------------------------------------
// Forward DWT level:  a[o] = sum_j h_lo[j] * x_sym[2o+1-j]   (d with h_hi)
// Tile o = tb + 16n + m  ->  D[m][n] = sum_k A[m][k] B[k][n]
//   A[m][k] = h[2m+15-k] (banded, K padded to 48), B[k][n] = x_sym[2tb+32n+k-14]
// Block = 128 threads = 4 waves, 4 tiles per group, grid-stride over groups.
// ---------------------------------------------------------------------------
__global__ __launch_bounds__(128) void dwt_fwd_wmma(
    const float* __restrict__ x, int n,
    float* __restrict__ outA, float* __restrict__ outD,
    int nOut, int numGroups)
{
  __shared__ float xS[2080];   // staged input span: 2064 used
  __shared__ float tLo[80];    // dec_lo padded table: tLo[j+32]=dec_lo[j]
  __shared__ float tHi[80];    // dec_hi padded table

  const int tid  = threadIdx.x;
  const int lane = tid & 31;
  const int w    = tid >> 5;

  for (int t = tid; t < 160; t += 128) {
    int which = t / 80, idx = t % 80, j = idx - 32;
    float lo = 0.f, hi = 0.f;
    if (j >= 0 && j < 16) {
      lo = DEC_LO[j];
      // dec_hi[j] = dec_lo[15-j] * (-1)^(15-j)
      hi = (j & 1) ? DEC_LO[15 - j] : -DEC_LO[15 - j];
    }
    if (which == 0) tLo[idx] = lo; else tHi[idx] = hi;
  }
  __syncthreads();

  // A operands: constant per lane for the whole kernel
  const int m     = lane & 15;
  const int cA    = (lane >> 4) << 1;   // lanes 16-31 hold K+2 columns
  const int half8 = (lane >> 4) << 3;
  v2f ALo[12], AHi[12];
#pragma unroll
  for (int kc = 0; kc < 12; ++kc) {
    const int k  = 4 * kc + cA;
    const int i0 = 2 * m + 15 - k + 32;       // table idx of h[2m+15-k]
    ALo[kc].x = tLo[i0]; ALo[kc].y = tLo[i0 - 1];
    AHi[kc].x = tHi[i0]; AHi[kc].y = tHi[i0 - 1];
  }

  for (int g = blockIdx.x; g < numGroups; g += gridDim.x) {
    const int base = 2048 * g - 14;           // staged span start (sym index)
    for (int s = tid; s < 2064; s += 128)
      xS[s] = x[symidx(base + s, n)];
    __syncthreads();

    const int tg = 4 * g + w;
    const int tb = tg << 8;
    if (tb < nOut) {
      v8f accA = {}; v8f accD = {};
#pragma unroll
      for (int kc = 0; kc < 12; ++kc) {
        const int s = (w << 9) + (m << 5) + 4 * kc + cA;  // 512w+32n+k
        v2f b = *(const v2f*)(xS + s);                     // 8B-aligned ds read
        accA = wmma4(ALo[kc], b, accA);
        accD = wmma4(AHi[kc], b, accD);
      }
      // C layout: lane holds 8 consecutive outputs -> two B128 stores
      const int ob = tb + (m << 4) + half8;
      if (ob + 8 <= nOut) {
        v4f a0 = {accA[0], accA[1], accA[2], accA[3]};
        v4f a1 = {accA[4], accA[5], accA[6], accA[7]};
        v4f d0 = {accD[0], accD[1], accD[2], accD[3]};
        v4f d1 = {accD[4], accD[5], accD[6], accD[7]};
        *(v4f*)(outA + ob)     = a0;  *(v4f*)(outA + ob + 4) = a1;
        *(v4f*)(outD + ob)     = d0;  *(v4f*)(outD + ob + 4) = d1;
      } else {
        for (int r = 0; r < 8; ++r)
          if (ob + r < nOut) { outA[ob + r] = accA[r]; outD[ob + r] = accD[r]; }
      }
    }
    __syncthreads();
  }
}

// ---------------------------------------------------------------------------
// Inverse DWT level: rec[t] = sum_i a[i]*rl[t+14-2i] + d[i]*rh[t+14-2i]
// Tile t = tb + 16n + m (tb even) -> i = tb/2 + 8n + q, q in [0,16)
//   A[m][q] = rl[m+14-2q] (valid range), B[q][n] = a[tb/2+8n+q] (zero-padded)
// ---------------------------------------------------------------------------
__global__ __launch_bounds__(128) void idwt_wmma(
    const float* __restrict__ aIn, int nA,
    const float* __restrict__ dIn, int nD,
    float* __restrict__ out, int outLen, int numGroups)
{
  __shared__ float aS[528];
  __shared__ float dS[528];
  __shared__ float tRL[48];   // tRL[j+16]=rec_lo[j]
  __shared__ float tRH[48];   // tRH[j+16]=rec_hi[j]

  const int tid  = threadIdx.x;
  const int lane = tid & 31;
  const int w    = tid >> 5;

  for (int t = tid; t < 96; t += 128) {
    int which = t / 48, idx = t % 48, j = idx - 16;
    float rl = 0.f, rh = 0.f;
    if (j >= 0 && j < 16) {
      rl = DEC_LO[15 - j];                         // rec_lo = reverse(dec_lo)
      rh = (j & 1) ? -DEC_LO[j] : DEC_LO[j];       // rec_hi[j]=dec_lo[j]*(-1)^j
    }
    if (which == 0) tRL[idx] = rl; else tRH[idx] = rh;
  }
  __syncthreads();

  const int m     = lane & 15;
  const int cA    = (lane >> 4) << 1;
  const int half8 = (lane >> 4) << 3;
  v2f ALo[4], AHi[4];
#pragma unroll
  for (int qc = 0; qc < 4; ++qc) {
    const int q  = 4 * qc + cA;
    const int j0 = m + 14 - 2 * q + 16;
    ALo[qc].x = tRL[j0]; ALo[qc].y = tRL[j0 - 2];
    AHi[qc].x = tRH[j0]; AHi[qc].y = tRH[j0 - 2];
  }

  for (int g = blockIdx.x; g < numGroups; g += gridDim.x) {
    const int base = 512 * g;
    for (int s = tid; s < 520; s += 128) {
      const int gi = base + s;
      aS[s] = (gi < nA) ? aIn[gi] : 0.0f;
      dS[s] = (gi < nD) ? dIn[gi] : 0.0f;
    }
    __syncthreads();

    const int tg = 4 * g + w;
    const int tb = tg << 8;
    if (tb < outLen) {
      v8f acc = {};
#pragma unroll
      for (int qc = 0; qc < 4; ++qc) {
        const int s = (w << 7) + (m << 3) + 4 * qc + cA;   // 128w+8n+q
        v2f ba = *(const v2f*)(aS + s);
        v2f bd = *(const v2f*)(dS + s);
        acc = wmma4(ALo[qc], ba, acc);
        acc = wmma4(AHi[qc], bd, acc);
      }
      const int ob = tb + (m << 4) + half8;
      if (ob + 8 <= outLen) {
        v4f r0 = {acc[0], acc[1], acc[2], acc[3]};
        v4f r1 = {acc[4], acc[5], acc[6], acc[7]};
        *(v4f*)(out + ob)     = r0;
        *(v4f*)(out + ob + 4) = r1;
      } else {
        for (int r = 0; r < 8; ++r)
          if (ob + r < outLen) out[ob + r] = acc[r];
      }
    }
    __syncthreads();
  }
}

// ----------------------- median / threshold machinery ----------------------
__global__ void zero_u32_kernel(unsigned* p, int n) {
  for (int i = blockIdx.x * blockDim.x + threadIdx.x; i < n;
       i += gridDim.x * blockDim.x) p[i] = 0u;
}

__global__ void hist_hi_kernel(const float* __restrict__ d, int n,
                               unsigned* __restrict__ hist) {
  for (int i = blockIdx.x * blockDim.x + threadIdx.x; i < n;
       i += gridDim.x * blockDim.x) {
    unsigned b = __float_as_uint(fabsf(d[i])) >> 16;
    atomicAdd(&hist[b], 1u);
  }
}

__global__ void select_hi_kernel(const unsigned* __restrict__ hist,
                                 unsigned targetRank, int* statsI) {
  __shared__ unsigned part[256];
  const int t = threadIdx.x;
  unsigned s = 0;
  for (int b = t * 256; b < t * 256 + 256; ++b) s += hist[b];
  part[t] = s;
  __syncthreads();
  if (t == 0) {
    unsigned long long cum = 0; int seg = 0;
    for (; seg < 256; ++seg) {
      if (cum + part[seg] > (unsigned long long)targetRank) break;
      cum += part[seg];
    }
    int b = seg * 256;
    for (;; ++b) {
      if (cum + hist[b] > (unsigned long long)targetRank) break;
      cum += hist[b];
    }
    statsI[0] = b;                                   // high-16 bin of median
    statsI[1] = (int)((unsigned long long)targetRank - cum);  // residual rank
  }
}

__global__ void hist_lo_kernel(const float* __restrict__ d, int n,
                               unsigned* __restrict__ hist,
                               const int* __restrict__ statsI) {
  const unsigned hi = (unsigned)statsI[0];
  for (int i = blockIdx.x * blockDim.x + threadIdx.x; i < n;
       i += gridDim.x * blockDim.x) {
    unsigned bits = __float_as_uint(fabsf(d[i]));
    if ((bits >> 16) == hi) atomicAdd(&hist[bits & 0xFFFFu], 1u);
  }
}

__global__ void select_lo_kernel(const unsigned* __restrict__ hist,
                                 int* statsI, float* statsF, float sqrt2logn) {
  __shared__ unsigned part[256];
  const int t = threadIdx.x;
  unsigned s = 0;
  for (int b = t * 256; b < t * 256 + 256; ++b) s += hist[b];
  part[t] = s;
  __syncthreads();
  if (t == 0) {
    const unsigned long long target = (unsigned long long)(unsigned)statsI[1];
    unsigned long long cum = 0; int seg = 0;
    for (; seg < 256; ++seg) {
      if (cum + part[seg] > target) break;
      cum += part[seg];
    }
    int b = seg * 256;
    for (;; ++b) {
      if (cum + hist[b] > target) break;
      cum += hist[b];
    }
    const unsigned bits = ((unsigned)statsI[0] << 16) | (unsigned)b;
    const float med   = __uint_as_float(bits);
    const float sigma = med / 0.6745f;
    const float lam   = sigma * sqrt2logn;
    statsF[2] = lam;
    statsF[3] = 0.85f * lam;
  }
}

__global__ void thresh_kernel(float* d, int n, const float* __restrict__ statsF) {
  const float lam = statsF[2];
  const float shr = statsF[3];
  for (int i = blockIdx.x * blockDim.x + threadIdx.x; i < n;
       i += gridDim.x * blockDim.x) {
    const float v = d[i];
    const float av = fabsf(v);
    d[i] = (av >= lam) ? copysignf(av - shr, v) : 0.0f;
  }
}

// ---------------------------------------------------------------------------
extern "C" void kernel_launch(void* const* d_in, const int* in_sizes, int n_in,
                              void* d_out, int out_size, void* d_ws, size_t ws_size,
                              hipStream_t stream) {
  (void)n_in; (void)out_size; (void)ws_size;
  const float* x = (const float*)d_in[0];
  const int n0 = in_sizes[0];                 // 16,777,216
  const int n1 = (n0 + 13) / 2 + 1;           // DWT level output length
  const int n2 = (n1 + 13) / 2 + 1;
  const int n3 = (n2 + 13) / 2 + 1;

  // workspace layout (floats); ~118 MB total
  float* ws = (float*)d_ws;
  size_t o = 0;
  float* d1 = ws + o; o += (size_t)n1;
  float* d2 = ws + o; o += (size_t)n2;
  float* d3 = ws + o; o += (size_t)n3;
  float* a1 = ws + o; o += (size_t)n1 + 1;    // reused as r2 (len <= n1+1)
  float* a2 = ws + o; o += (size_t)n2 + 1;    // reused as r3 (len <= n2+1)
  float* a3 = ws + o; o += (size_t)n3;
  unsigned* hist0 = (unsigned*)(ws + o); o += 65536;
  unsigned* hist1 = (unsigned*)(ws + o); o += 65536;
  int*   statsI = (int*)(ws + o);
  float* statsF = (float*)(ws + o); o += 16;

  auto launch_fwd = [&](const float* in, int n, float* oa, float* od, int nout) {
    const int tiles  = (nout + 255) / 256;
    const int groups = (tiles + 3) / 4;
    dwt_fwd_wmma<<<groups, 128, 0, stream>>>(in, n, oa, od, nout, groups);
  };
  auto launch_idwt = [&](const float* a, int na, const float* d, int nd,
                         float* outp, int outLen) {
    const int tiles  = (outLen + 255) / 256;
    const int groups = (tiles + 3) / 4;
    idwt_wmma<<<groups, 128, 0, stream>>>(a, na, d, nd, outp, outLen, groups);
  };

  // forward 3-level DWT
  launch_fwd(x,  n0, a1, d1, n1);
  launch_fwd(a1, n1, a2, d2, n2);
  launch_fwd(a2, n2, a3, d3, n3);

  // exact median of |d1| via 2-level 16-bit radix select (n1 odd -> one rank)
  zero_u32_kernel<<<512, 256, 0, stream>>>(hist0, 2 * 65536 + 16);
  hist_hi_kernel<<<2048, 256, 0, stream>>>(d1, n1, hist0);
  const unsigned rank = (unsigned)((n1 - 1) / 2);
  select_hi_kernel<<<1, 256, 0, stream>>>(hist0, rank, statsI);
  hist_lo_kernel<<<2048, 256, 0, stream>>>(d1, n1, hist1, statsI);
  const float s2ln = (float)sqrt(2.0 * log((double)n0));
  select_lo_kernel<<<1, 256, 0, stream>>>(hist1, statsI, statsF, s2ln);

  // soft-shrink detail bands in place
  thresh_kernel<<<2048, 256, 0, stream>>>(d1, n1, statsF);
  thresh_kernel<<<1024, 256, 0, stream>>>(d2, n2, statsF);
  thresh_kernel<<<512,  256, 0, stream>>>(d3, n3, statsF);

  // inverse 3-level DWT (a truncated to len(d) each level)
  const int o3 = 2 * n3 - 14;
  launch_idwt(a3, n3, d3, n3, a2 /*r3*/, o3);
  launch_idwt(a2, n2, d2, n2, a1 /*r2*/, 2 * n2 - 14);
  launch_idwt(a1, n1, d1, n1, (float*)d_out, n0);   // 2*n1-14 == n0
}